// SAGE_46823733461095
// MI455X (gfx1250) — compile-verified
//
#include <hip/hip_runtime.h>
#include <hip/hip_bf16.h>

#define NNODES 100000
#define NEDGES 1200000
#define NF 64
#define NH 64
#define NC 16
#define BN_EPS 1e-5f
#define L2_EPS 1e-12f

typedef __attribute__((ext_vector_type(2))) float v2f;
typedef __attribute__((ext_vector_type(8))) float v8f;

// D(16x16,f32) += A(16x4,f32) x B(4x16,f32)
__device__ __forceinline__ v8f wmma4(v2f a, v2f b, v8f c) {
    return __builtin_amdgcn_wmma_f32_16x16x4_f32(
        /*neg_a=*/false, a, /*neg_b=*/false, b,
        /*c_mod=*/(short)0, c, /*reuse_a=*/false, /*reuse_b=*/false);
}

// ---------------- kernel 1: zero the accumulation buffers ----------------
__global__ void k_zero(float* p, long long n) {
    long long i = (long long)blockIdx.x * blockDim.x + threadIdx.x;
    long long stride = (long long)gridDim.x * blockDim.x;
    for (; i < n; i += stride) p[i] = 0.0f;
}

// ---------------- kernel 2: edge scatter (one wave32 per edge) ----------------
__global__ void k_scatter(const int* __restrict__ ei, const float* __restrict__ x,
                          float* __restrict__ sums, float* __restrict__ cnt) {
    long long t = (long long)blockIdx.x * blockDim.x + threadIdx.x;
    int e = (int)(t >> 5);
    int lane = (int)(t & 31);
    if (e >= NEDGES) return;
    int src = ei[e];
    int dst = ei[NEDGES + e];
    const v2f v = *reinterpret_cast<const v2f*>(x + (size_t)src * NF + lane * 2);
    atomicAdd(&sums[(size_t)dst * NF + lane * 2 + 0], v.x);
    atomicAdd(&sums[(size_t)dst * NF + lane * 2 + 1], v.y);
    if (lane == 0) atomicAdd(&cnt[dst], 1.0f);
}

// ---------------- kernel 3: sums -> mean (in place) ----------------
__global__ void k_mean(float* __restrict__ sums, const float* __restrict__ cnt) {
    long long i = ((long long)blockIdx.x * blockDim.x + threadIdx.x) * 2;
    if (i >= (long long)NNODES * NF) return;
    float c = fmaxf(cnt[i >> 6], 1.0f);
    float inv = 1.0f / c;
    v2f v = *reinterpret_cast<v2f*>(sums + i);
    v.x *= inv; v.y *= inv;
    *reinterpret_cast<v2f*>(sums + i) = v;
}

// ---------------- kernel 4: h = mean @ Wl^T + b_l + x @ Wr^T (f32 WMMA) ----
// block = 128 threads = 4 waves; block handles 16 rows, wave w handles 16 cols.
__global__ void k_gemm_h(const float* __restrict__ mean, const float* __restrict__ x,
                         const float* __restrict__ Wl, const float* __restrict__ bl,
                         const float* __restrict__ Wr, float* __restrict__ h) {
    const int lane = threadIdx.x & 31;
    const int wave = threadIdx.x >> 5;
    const int row_base = blockIdx.x * 16;
    const int col_base = wave * 16;
    const int lo = lane & 15;         // M index (A) / N index (B,C)
    const int hi = lane >> 4;         // K sub-offset selector
    const int koff = hi * 2;

    const float* arow_m = mean + (size_t)(row_base + lo) * NF;
    const float* arow_x = x    + (size_t)(row_base + lo) * NF;
    const float* brow_l = Wl   + (size_t)(col_base + lo) * NF; // B[k][n] = Wl[n][k]
    const float* brow_r = Wr   + (size_t)(col_base + lo) * NF;

    v8f c;
    float bias = bl[col_base + lo];   // bias depends only on N column
    #pragma unroll
    for (int j = 0; j < 8; ++j) c[j] = bias;

    #pragma unroll
    for (int k0 = 0; k0 < NF; k0 += 4) {
        v2f a = *reinterpret_cast<const v2f*>(arow_m + k0 + koff);
        v2f b = *reinterpret_cast<const v2f*>(brow_l + k0 + koff);
        c = wmma4(a, b, c);
    }
    #pragma unroll
    for (int k0 = 0; k0 < NF; k0 += 4) {
        v2f a = *reinterpret_cast<const v2f*>(arow_x + k0 + koff);
        v2f b = *reinterpret_cast<const v2f*>(brow_r + k0 + koff);
        c = wmma4(a, b, c);
    }

    // C/D layout: VGPR j -> (M = j + hi*8, N = lo)
    #pragma unroll
    for (int j = 0; j < 8; ++j)
        h[(size_t)(row_base + hi * 8 + j) * NH + col_base + lo] = c[j];
}

// ---------------- kernel 5: L2 normalize rows + ReLU + BN stats ----------------
// block = 256 = 8 waves; each wave owns one row per iteration (64 f32, 2/lane)
__global__ void k_norm_stats(float* __restrict__ h, float* __restrict__ bnsum,
                             float* __restrict__ bnsq) {
    __shared__ float ssum[NH];
    __shared__ float ssq[NH];
    const int lane = threadIdx.x & 31;
    const int wave = threadIdx.x >> 5;
    if (threadIdx.x < NH) { ssum[threadIdx.x] = 0.0f; ssq[threadIdx.x] = 0.0f; }
    __syncthreads();

    float ls0 = 0.0f, ls1 = 0.0f, lq0 = 0.0f, lq1 = 0.0f;
    for (int row = blockIdx.x * 8 + wave; row < NNODES; row += gridDim.x * 8) {
        float* p = h + (size_t)row * NH + lane * 2;
        v2f v = *reinterpret_cast<v2f*>(p);
        float sq = v.x * v.x + v.y * v.y;
        #pragma unroll
        for (int off = 16; off > 0; off >>= 1) sq += __shfl_xor(sq, off, 32);
        float scale = 1.0f / fmaxf(sqrtf(sq), L2_EPS);
        v.x = fmaxf(v.x * scale, 0.0f);
        v.y = fmaxf(v.y * scale, 0.0f);
        *reinterpret_cast<v2f*>(p) = v;
        ls0 += v.x; ls1 += v.y;
        lq0 += v.x * v.x; lq1 += v.y * v.y;
    }
    atomicAdd(&ssum[lane * 2 + 0], ls0);
    atomicAdd(&ssum[lane * 2 + 1], ls1);
    atomicAdd(&ssq[lane * 2 + 0], lq0);
    atomicAdd(&ssq[lane * 2 + 1], lq1);
    __syncthreads();
    if (threadIdx.x < NH) {
        atomicAdd(&bnsum[threadIdx.x], ssum[threadIdx.x]);
        atomicAdd(&bnsq[threadIdx.x], ssq[threadIdx.x]);
    }
}

// ---------------- kernel 6: fold BN into fc weights ----------------
// Weff[c][k] = Wfc[c][k]*gamma[k]*invstd[k]
// beff[c]    = bfc[c] + sum_k Wfc[c][k]*(beta[k] - mu[k]*gamma[k]*invstd[k])
__global__ void k_fold(const float* __restrict__ bnsum, const float* __restrict__ bnsq,
                       const float* __restrict__ gamma, const float* __restrict__ beta,
                       const float* __restrict__ Wfc, const float* __restrict__ bfc,
                       float* __restrict__ Weff, float* __restrict__ beff) {
    __shared__ float red[NC][NH];
    const int k = threadIdx.x;   // 0..63
    const int c = threadIdx.y;   // 0..15
    const float invN = 1.0f / (float)NNODES;
    float mu = bnsum[k] * invN;
    float var = bnsq[k] * invN - mu * mu;
    float g = gamma[k] * rsqrtf(var + BN_EPS);
    float w = Wfc[c * NH + k];
    Weff[c * NH + k] = w * g;
    red[c][k] = w * (beta[k] - mu * g);
    __syncthreads();
    if (k == 0) {
        float s = bfc[c];
        #pragma unroll
        for (int i = 0; i < NH; ++i) s += red[c][i];
        beff[c] = s;
    }
}

// ---------------- kernel 7: out = h @ Weff^T + beff (f32 WMMA) ----------------
// block = 256 = 8 waves; each wave computes one 16x16 output tile (NC == 16)
__global__ void k_gemm_out(const float* __restrict__ h, const float* __restrict__ Weff,
                           const float* __restrict__ beff, float* __restrict__ out) {
    const int lane = threadIdx.x & 31;
    const int wave = threadIdx.x >> 5;
    const int tile = blockIdx.x * 8 + wave;
    if (tile >= NNODES / 16) return;      // uniform per wave -> EXEC all-ones inside
    const int row_base = tile * 16;
    const int lo = lane & 15;
    const int hi = lane >> 4;
    const int koff = hi * 2;

    const float* arow = h    + (size_t)(row_base + lo) * NH;
    const float* brow = Weff + (size_t)lo * NH;   // B[k][n] = Weff[n][k]

    v8f c;
    float bias = beff[lo];
    #pragma unroll
    for (int j = 0; j < 8; ++j) c[j] = bias;

    #pragma unroll
    for (int k0 = 0; k0 < NH; k0 += 4) {
        v2f a = *reinterpret_cast<const v2f*>(arow + k0 + koff);
        v2f b = *reinterpret_cast<const v2f*>(brow + k0 + koff);
        c = wmma4(a, b, c);
    }
    #pragma unroll
    for (int j = 0; j < 8; ++j)
        out[(size_t)(row_base + hi * 8 + j) * NC + lo] = c[j];
}

extern "C" void kernel_launch(void* const* d_in, const int* in_sizes, int n_in,
                              void* d_out, int out_size, void* d_ws, size_t ws_size,
                              hipStream_t stream) {
    const int*   ei    = (const int*)  d_in[0];   // [2, E]
    const float* x     = (const float*)d_in[1];   // [N, 64]
    const float* Wl    = (const float*)d_in[2];   // [64, 64]
    const float* bl    = (const float*)d_in[3];   // [64]
    const float* Wr    = (const float*)d_in[4];   // [64, 64]
    const float* gamma = (const float*)d_in[5];   // [64]
    const float* beta  = (const float*)d_in[6];   // [64]
    const float* Wfc   = (const float*)d_in[7];   // [16, 64]
    const float* bfc   = (const float*)d_in[8];   // [16]
    float* out = (float*)d_out;                   // [N, 16]

    float* ws = (float*)d_ws;
    size_t off = 0;
    float* sums  = ws + off; off += (size_t)NNODES * NF;   // reused as mean
    float* cnt   = ws + off; off += NNODES;
    float* hbuf  = ws + off; off += (size_t)NNODES * NH;
    float* bnsum = ws + off; off += NH;
    float* bnsq  = ws + off; off += NH;
    float* Weff  = ws + off; off += NC * NH;
    float* beff  = ws + off; off += NC;

    // 1) zero accumulators (sums, cnt, bnsum, bnsq are contiguous except bn*)
    long long nzero = (long long)NNODES * NF + NNODES;      // sums + cnt
    k_zero<<<4096, 256, 0, stream>>>(sums, nzero);
    k_zero<<<1, 256, 0, stream>>>(bnsum, 2 * NH);           // bnsum + bnsq contiguous

    // 2) scatter-add over edges: one wave per edge
    {
        long long threads = (long long)NEDGES * 32;
        int blocks = (int)((threads + 255) / 256);
        k_scatter<<<blocks, 256, 0, stream>>>(ei, x, sums, cnt);
    }

    // 3) mean = sums / max(cnt, 1)
    {
        long long elems2 = (long long)NNODES * NF / 2;
        int blocks = (int)((elems2 + 255) / 256);
        k_mean<<<blocks, 256, 0, stream>>>(sums, cnt);
    }

    // 4) h = mean @ Wl^T + bl + x @ Wr^T   (N = 6250 * 16 exactly)
    k_gemm_h<<<NNODES / 16, 128, 0, stream>>>(sums, x, Wl, bl, Wr, hbuf);

    // 5) row L2-normalize + ReLU + BN stats
    k_norm_stats<<<1024, 256, 0, stream>>>(hbuf, bnsum, bnsq);

    // 6) fold BN affine into fc weights
    k_fold<<<1, dim3(NH, NC), 0, stream>>>(bnsum, bnsq, gamma, beta, Wfc, bfc, Weff, beff);

    // 7) out = h @ Weff^T + beff
    {
        int tiles = NNODES / 16;
        int blocks = (tiles + 7) / 8;
        k_gemm_out<<<blocks, 256, 0, stream>>>(hbuf, Weff, beff, out);
    }
    (void)in_sizes; (void)n_in; (void)out_size; (void)ws_size;
}